// GraphEncoder_13245679141086
// MI455X (gfx1250) — compile-verified
//
#include <hip/hip_runtime.h>
#include <math.h>

#define NNODES 50000
#define NEDGES 800000
#define DIN0   128
#define NH     8
#define CHD    32
#define HCDIM  256
#define EDIM   16
#define OUTDIM 256
#define NLAYER 4
#define EL     400000     // edges used per layer (every other edge)
#define BNEPS  1e-5f
#define INV_SQRT_C 0.17677669529663687f

typedef __attribute__((ext_vector_type(2))) float v2f;
typedef __attribute__((ext_vector_type(8))) float v8f;

__device__ __forceinline__ v2f ld2(const float* __restrict__ p, int o) {
  v2f r; r.x = p[o]; r.y = p[o + 1]; return r;
}

// ---------------------------------------------------------------------------
// Fused 4-way fp32 WMMA GEMM (software-pipelined, DIN compile-time):
//   for m in {q,k,v,skip}:  Ym[n, 256] = X[n, DIN] @ Wm[256, DIN]^T + bm
// One wave computes one 16x16 tile for all four weight matrices; the
// A-fragment is loaded once per K-step and feeds 4 WMMAs. Next K-step's
// fragments are prefetched before the current WMMAs issue so global-load
// latency overlaps the matrix pipe instead of serializing on loadcnt==0.
// ---------------------------------------------------------------------------
template <int DIN>
__global__ __launch_bounds__(256) void gemm4_bias_wmma(
    const float* __restrict__ X,
    const float* __restrict__ W0, const float* __restrict__ b0,
    const float* __restrict__ W1, const float* __restrict__ b1,
    const float* __restrict__ W2, const float* __restrict__ b2,
    const float* __restrict__ W3, const float* __restrict__ b3,
    float* __restrict__ Y0, float* __restrict__ Y1,
    float* __restrict__ Y2, float* __restrict__ Y3,
    int nrows) {
  const int colTiles = HCDIM >> 4;
  const int totalTiles = (nrows >> 4) * colTiles;
  int wid  = threadIdx.x >> 5;
  int lane = threadIdx.x & 31;
  int tile = blockIdx.x * (blockDim.x >> 5) + wid;
  if (tile >= totalTiles) return;
  int rowTile = tile / colTiles;
  int colTile = tile - rowTile * colTiles;

  int half  = lane >> 4;      // 0: K pair {0,1}, 1: K pair {2,3}
  int idx16 = lane & 15;

  const size_t xoff = (size_t)((rowTile << 4) + idx16) * DIN;
  const size_t woff = (size_t)((colTile << 4) + idx16) * DIN;
  const float* xrow = X  + xoff;
  const float* w0r  = W0 + woff;
  const float* w1r  = W1 + woff;
  const float* w2r  = W2 + woff;
  const float* w3r  = W3 + woff;

  v8f acc0 = {}, acc1 = {}, acc2 = {}, acc3 = {};

  // Prologue: fragments for K-step 0.
  int ka = 2 * half;
  v2f a  = ld2(xrow, ka);
  v2f f0 = ld2(w0r, ka), f1 = ld2(w1r, ka), f2 = ld2(w2r, ka), f3 = ld2(w3r, ka);

#pragma unroll
  for (int k0 = 0; k0 < DIN - 4; k0 += 4) {
    int kn = k0 + 4 + 2 * half;
    // Prefetch next K-step while current WMMAs execute.
    v2f an  = ld2(xrow, kn);
    v2f g0  = ld2(w0r, kn), g1 = ld2(w1r, kn), g2 = ld2(w2r, kn), g3 = ld2(w3r, kn);
    acc0 = __builtin_amdgcn_wmma_f32_16x16x4_f32(false, a, false, f0, (short)0, acc0, false, false);
    acc1 = __builtin_amdgcn_wmma_f32_16x16x4_f32(false, a, false, f1, (short)0, acc1, false, false);
    acc2 = __builtin_amdgcn_wmma_f32_16x16x4_f32(false, a, false, f2, (short)0, acc2, false, false);
    acc3 = __builtin_amdgcn_wmma_f32_16x16x4_f32(false, a, false, f3, (short)0, acc3, false, false);
    a = an; f0 = g0; f1 = g1; f2 = g2; f3 = g3;
  }
  // Epilogue K-step.
  acc0 = __builtin_amdgcn_wmma_f32_16x16x4_f32(false, a, false, f0, (short)0, acc0, false, false);
  acc1 = __builtin_amdgcn_wmma_f32_16x16x4_f32(false, a, false, f1, (short)0, acc1, false, false);
  acc2 = __builtin_amdgcn_wmma_f32_16x16x4_f32(false, a, false, f2, (short)0, acc2, false, false);
  acc3 = __builtin_amdgcn_wmma_f32_16x16x4_f32(false, a, false, f3, (short)0, acc3, false, false);

  // C/D layout: VGPR v holds row (v + 8*half), lane idx16 holds column idx16.
  int rbase = (rowTile << 4) + 8 * half;
  int c     = (colTile << 4) + idx16;
  float bv0 = b0[c], bv1 = b1[c], bv2 = b2[c], bv3 = b3[c];
#pragma unroll
  for (int v = 0; v < 8; ++v) {
    size_t o = (size_t)(rbase + v) * HCDIM + c;
    Y0[o] = acc0[v] + bv0;
    Y1[o] = acc1[v] + bv1;
    Y2[o] = acc2[v] + bv2;
    Y3[o] = acc3[v] + bv3;
  }
}

// ---------------------------------------------------------------------------
// Single fp32 WMMA GEMM (output head), same pipelining.
// ---------------------------------------------------------------------------
template <int DIN>
__global__ __launch_bounds__(256) void gemm_bias_wmma(
    const float* __restrict__ X, const float* __restrict__ W,
    const float* __restrict__ bias, float* __restrict__ Y,
    int nrows, int dout) {
  const int colTiles = dout >> 4;
  const int totalTiles = (nrows >> 4) * colTiles;
  int wid  = threadIdx.x >> 5;
  int lane = threadIdx.x & 31;
  int tile = blockIdx.x * (blockDim.x >> 5) + wid;
  if (tile >= totalTiles) return;
  int rowTile = tile / colTiles;
  int colTile = tile - rowTile * colTiles;

  int half  = lane >> 4;
  int idx16 = lane & 15;

  const float* xrow = X + (size_t)((rowTile << 4) + idx16) * DIN;
  const float* wrow = W + (size_t)((colTile << 4) + idx16) * DIN;

  v8f acc = {};
  int ka = 2 * half;
  v2f a = ld2(xrow, ka), b = ld2(wrow, ka);
#pragma unroll
  for (int k0 = 0; k0 < DIN - 4; k0 += 4) {
    int kn = k0 + 4 + 2 * half;
    v2f an = ld2(xrow, kn), bn = ld2(wrow, kn);
    acc = __builtin_amdgcn_wmma_f32_16x16x4_f32(false, a, false, b, (short)0, acc, false, false);
    a = an; b = bn;
  }
  acc = __builtin_amdgcn_wmma_f32_16x16x4_f32(false, a, false, b, (short)0, acc, false, false);

  int rbase = (rowTile << 4) + 8 * half;
  int c     = (colTile << 4) + idx16;
  float bv  = bias[c];
#pragma unroll
  for (int v = 0; v < 8; ++v)
    Y[(size_t)(rbase + v) * dout + c] = acc[v] + bv;
}

// ---------------------------------------------------------------------------
// float atomic max via int/uint ordering trick (init with -inf)
// ---------------------------------------------------------------------------
__device__ __forceinline__ void atomicMaxF(float* addr, float val) {
  if (val >= 0.0f) atomicMax((int*)addr, __float_as_int(val));
  else             atomicMin((unsigned int*)addr, __float_as_uint(val));
}

__global__ void init_softmax_state(float* __restrict__ m, float* __restrict__ denom, int n) {
  int i = blockIdx.x * blockDim.x + threadIdx.x;
  if (i < n) { m[i] = -INFINITY; denom[i] = 0.0f; }
}

// ---------------------------------------------------------------------------
// Per-edge attention logits: alpha[e,h] = q[dst]·(k[src]+e)/sqrt(C); seg-max.
// One wave per edge; lane = channel within head; edge proj recomputed (16 FMA).
// ---------------------------------------------------------------------------
__global__ __launch_bounds__(256) void edge_alpha_kernel(
    const int* __restrict__ ei, const float* __restrict__ ea,
    const float* __restrict__ We, const float* __restrict__ q,
    const float* __restrict__ k, float* __restrict__ alpha,
    float* __restrict__ m, int parity) {
  int wid  = threadIdx.x >> 5;
  int lane = threadIdx.x & 31;
  int e = blockIdx.x * (blockDim.x >> 5) + wid;
  if (e >= EL) return;
  int ge  = 2 * e + parity;
  int src = ei[ge];
  int dst = ei[NEDGES + ge];

  float ead[EDIM];
#pragma unroll
  for (int d = 0; d < EDIM; ++d) ead[d] = ea[(size_t)ge * EDIM + d];

  const float* qrow = q + (size_t)dst * HCDIM;
  const float* krow = k + (size_t)src * HCDIM;

#pragma unroll
  for (int h = 0; h < NH; ++h) {
    int hc = h * CHD + lane;
    float ev = 0.0f;
#pragma unroll
    for (int d = 0; d < EDIM; ++d) ev = fmaf(ead[d], We[hc * EDIM + d], ev);
    float p = qrow[hc] * (krow[hc] + ev);
#pragma unroll
    for (int off = 16; off > 0; off >>= 1) p += __shfl_xor(p, off, 32);
    if (lane == 0) {
      float a = p * INV_SQRT_C;
      alpha[(size_t)e * NH + h] = a;
      atomicMaxF(&m[dst * NH + h], a);
    }
  }
}

// ex = exp(alpha - m[dst]); denom[dst] += ex   (ex overwrites alpha in place)
__global__ void edge_softmax_kernel(const int* __restrict__ ei,
                                    float* __restrict__ alpha,
                                    const float* __restrict__ m,
                                    float* __restrict__ denom, int parity) {
  int idx = blockIdx.x * blockDim.x + threadIdx.x;
  if (idx >= EL * NH) return;
  int e = idx / NH, h = idx - e * NH;
  int ge  = 2 * e + parity;
  int dst = ei[NEDGES + ge];
  float ex = __expf(alpha[idx] - m[dst * NH + h]);
  alpha[idx] = ex;
  atomicAdd(&denom[dst * NH + h], ex);
}

// agg[dst] += (v[src] + e) * attn ; agg pre-initialized with skip projection
__global__ __launch_bounds__(256) void edge_message_kernel(
    const int* __restrict__ ei, const float* __restrict__ ea,
    const float* __restrict__ We, const float* __restrict__ vbuf,
    const float* __restrict__ exv, const float* __restrict__ denom,
    float* __restrict__ agg, int parity) {
  int wid  = threadIdx.x >> 5;
  int lane = threadIdx.x & 31;
  int e = blockIdx.x * (blockDim.x >> 5) + wid;
  if (e >= EL) return;
  int ge  = 2 * e + parity;
  int src = ei[ge];
  int dst = ei[NEDGES + ge];

  float ead[EDIM];
#pragma unroll
  for (int d = 0; d < EDIM; ++d) ead[d] = ea[(size_t)ge * EDIM + d];

  const float* vrow = vbuf + (size_t)src * HCDIM;
  float* aggrow = agg + (size_t)dst * HCDIM;

#pragma unroll
  for (int h = 0; h < NH; ++h) {
    float attn = exv[(size_t)e * NH + h] / fmaxf(denom[dst * NH + h], 1e-16f);
    int hc = h * CHD + lane;
    float ev = 0.0f;
#pragma unroll
    for (int d = 0; d < EDIM; ++d) ev = fmaf(ead[d], We[hc * EDIM + d], ev);
    atomicAdd(&aggrow[hc], (vrow[hc] + ev) * attn);
  }
}

// ---------------------------------------------------------------------------
// BatchNorm (biased var) + ReLU
// ---------------------------------------------------------------------------
__global__ void zero_stats(float* __restrict__ S1, float* __restrict__ S2) {
  int t = threadIdx.x;
  S1[t] = 0.0f; S2[t] = 0.0f;
}

__global__ void bn_partial(const float* __restrict__ X, float* __restrict__ S1,
                           float* __restrict__ S2, int rowsPer) {
  int col = threadIdx.x;                 // 0..255, coalesced across columns
  int r0 = blockIdx.x * rowsPer;
  int r1 = min(r0 + rowsPer, NNODES);
  float s1 = 0.0f, s2 = 0.0f;
  for (int r = r0; r < r1; ++r) {
    float x = X[(size_t)r * HCDIM + col];
    s1 += x; s2 += x * x;
  }
  atomicAdd(&S1[col], s1);
  atomicAdd(&S2[col], s2);
}

__global__ void bn_finalize(const float* __restrict__ S1, const float* __restrict__ S2,
                            const float* __restrict__ gamma, const float* __restrict__ beta,
                            float* __restrict__ scale, float* __restrict__ shift) {
  int t = threadIdx.x;
  float mu  = S1[t] * (1.0f / NNODES);
  float var = S2[t] * (1.0f / NNODES) - mu * mu;
  float g   = gamma[t] * rsqrtf(var + BNEPS);
  scale[t] = g;
  shift[t] = beta[t] - mu * g;
}

__global__ void bn_apply_relu(const float* __restrict__ X, const float* __restrict__ scale,
                              const float* __restrict__ shift, float* __restrict__ Y) {
  int idx = blockIdx.x * blockDim.x + threadIdx.x;   // NNODES*256 threads
  int col = idx & (HCDIM - 1);
  float y = fmaf(X[idx], scale[col], shift[col]);
  Y[idx] = fmaxf(y, 0.0f);
}

// ---------------------------------------------------------------------------
// Host orchestration
// ---------------------------------------------------------------------------
extern "C" void kernel_launch(void* const* d_in, const int* in_sizes, int n_in,
                              void* d_out, int out_size, void* d_ws, size_t ws_size,
                              hipStream_t stream) {
  const float* x_in = (const float*)d_in[0];
  const int*   ei   = (const int*)d_in[1];
  const float* ea   = (const float*)d_in[2];
  auto P = [&](int i) { return (const float*)d_in[i]; };

  const size_t NF = (size_t)NNODES * HCDIM;      // 12.8M floats
  float* ws    = (float*)d_ws;
  float* xcur  = ws;                             // activations between layers
  float* qbuf  = ws + 1 * NF;
  float* kbuf  = ws + 2 * NF;
  float* vbuf  = ws + 3 * NF;
  float* agg   = ws + 4 * NF;
  float* alpha = ws + 5 * NF;                    // EL*NH = 3.2M (alpha, then ex)
  float* mbuf  = alpha + (size_t)EL * NH;        // NNODES*NH
  float* denom = mbuf + (size_t)NNODES * NH;
  float* S1    = denom + (size_t)NNODES * NH;
  float* S2    = S1 + HCDIM;
  float* scl   = S2 + HCDIM;
  float* shf   = scl + HCDIM;

  const int gemmBlocks  = ((NNODES / 16) * (HCDIM / 16) + 7) / 8;  // 8 waves/block
  const int edgeBlocks  = (EL + 7) / 8;                            // 8 edges/block
  const int smBlocks    = (EL * NH + 255) / 256;
  const int initBlocks  = (NNODES * NH + 255) / 256;
  const int applyBlocks = (int)(NF / 256);
  const int bnChunks    = 200;                   // 250 rows per chunk
  const int bnRowsPer   = (NNODES + bnChunks - 1) / bnChunks;

  auto run_bn = [&](const float* X, const float* gamma, const float* beta, float* Y) {
    zero_stats<<<1, HCDIM, 0, stream>>>(S1, S2);
    bn_partial<<<bnChunks, HCDIM, 0, stream>>>(X, S1, S2, bnRowsPer);
    bn_finalize<<<1, HCDIM, 0, stream>>>(S1, S2, gamma, beta, scl, shf);
    bn_apply_relu<<<applyBlocks, 256, 0, stream>>>(X, scl, shf, Y);
  };

  for (int l = 0; l < NLAYER; ++l) {
    const int base = 3 + 11 * l;
    const float* Wq = P(base + 0); const float* bq = P(base + 1);
    const float* Wk = P(base + 2); const float* bk = P(base + 3);
    const float* Wv = P(base + 4); const float* bv = P(base + 5);
    const float* We = P(base + 6);
    const float* Ws = P(base + 7); const float* bs = P(base + 8);
    const float* gamma = P(base + 9); const float* beta = P(base + 10);

    const float* X = (l == 0) ? x_in : xcur;
    const int parity = l & 1;

    // Fused Q/K/V/Skip projections: X read once, 4 WMMAs per A-fragment.
    if (l == 0) {
      gemm4_bias_wmma<DIN0><<<gemmBlocks, 256, 0, stream>>>(
          X, Wq, bq, Wk, bk, Wv, bv, Ws, bs, qbuf, kbuf, vbuf, agg, NNODES);
    } else {
      gemm4_bias_wmma<HCDIM><<<gemmBlocks, 256, 0, stream>>>(
          X, Wq, bq, Wk, bk, Wv, bv, Ws, bs, qbuf, kbuf, vbuf, agg, NNODES);
    }

    init_softmax_state<<<initBlocks, 256, 0, stream>>>(mbuf, denom, NNODES * NH);
    edge_alpha_kernel<<<edgeBlocks, 256, 0, stream>>>(ei, ea, We, qbuf, kbuf, alpha, mbuf, parity);
    edge_softmax_kernel<<<smBlocks, 256, 0, stream>>>(ei, alpha, mbuf, denom, parity);
    edge_message_kernel<<<edgeBlocks, 256, 0, stream>>>(ei, ea, We, vbuf, alpha, denom, agg, parity);

    run_bn(agg, gamma, beta, xcur);
  }

  // Output head: GEMM + BN + ReLU -> d_out
  const float* Wo = P(47); const float* bo = P(48);
  const float* go = P(49); const float* be = P(50);
  gemm_bias_wmma<HCDIM><<<gemmBlocks, 256, 0, stream>>>(xcur, Wo, bo, qbuf, NNODES, OUTDIM);
  run_bn(qbuf, go, be, (float*)d_out);
}